// _Dip_Module_1692217115262
// MI455X (gfx1250) — compile-verified
//
#include <hip/hip_runtime.h>
#include <hip/hip_bf16.h>
#include <math.h>

// ---------------------------------------------------------------------------
// Types for WMMA fragments (CDNA5 / gfx1250, wave32)
// ---------------------------------------------------------------------------
typedef __attribute__((ext_vector_type(2))) float v2f;  // A/B frag of 16x16x4 f32
typedef __attribute__((ext_vector_type(8))) float v8f;  // C/D frag (16x16 f32)

static constexpr int DCONST = 128;   // feature dim, fixed by reference setup_inputs

// ---------------------------------------------------------------------------
// Kernel 1: projection matvec  y[i] = dot(X[i,:], p)  via V_WMMA_F32_16X16X4_F32
//   - each wave owns a 16-row tile of X
//   - A[m,k] = X[r0+m, kb+k]   (16x4 fp32; lanes 0-15 hold K={0,1}, lanes 16-31
//     hold K={2,3} per the ISA 32-bit A layout)
//   - B[k,n] = p[kb+k] replicated across all 16 columns -> C[m,n] = y[r0+m] ∀n
//   - out-of-range rows CLAMP their pointer to row N-1 (C rows are independent,
//     garbage rows are overwritten with +INF at store) -> branch-free K-loop,
//     EXEC all-ones around every WMMA
//   - K-loop fully unrolled (D=128 -> 32 WMMAs, immediate-offset b64 loads)
// ---------------------------------------------------------------------------
__global__ __launch_bounds__(128) void matvec_wmma(
    const float* __restrict__ X, const float* __restrict__ P,
    const int* __restrict__ pidx, float* __restrict__ y,
    int N, int Npad)
{
  const int lane = threadIdx.x & 31;
  const int wave = threadIdx.x >> 5;
  const int r0 = (blockIdx.x * 4 + wave) * 16;
  if (r0 >= Npad) return;                     // wave-uniform guard

  const float* p = P + (size_t)(*pidx) * DCONST;   // device-side scalar index

  const int half = lane >> 4;                 // 0: K pair {0,1}; 1: K pair {2,3}
  const int m    = lane & 15;                 // row within tile
  const int row  = r0 + m;
  const int crow = (row < N) ? row : (N - 1); // clamp: keeps loads in-bounds
  const float* xrow = X + (size_t)crow * DCONST + 2 * half;
  const float* prow = p + 2 * half;

  // speculative prefetch of the tile 64 rows ahead (silently dropped if OOB)
  __builtin_prefetch(xrow + 64 * DCONST, 0, 1);

  v8f c = {0.f, 0.f, 0.f, 0.f, 0.f, 0.f, 0.f, 0.f};

#pragma unroll
  for (int kb = 0; kb < DCONST; kb += 4) {
    const v2f a = *(const v2f*)(xrow + kb);   // global_load_b64, imm offset
    const v2f b = *(const v2f*)(prow + kb);   // p chunk (L0/L2 resident)
    c = __builtin_amdgcn_wmma_f32_16x16x4_f32(
        false, a, false, b, (short)0, c, false, false);
  }

  // C layout: VGPR v of lane L holds C[M = v + 8*(L>>4)][Nn = L&15].
  // All columns equal -> extract from Nn==0 holders (lanes 0 and 16).
  if (m == 0) {
    const int mbase = r0 + 8 * half;
#pragma unroll
    for (int v = 0; v < 8; ++v) {
      const int rr = mbase + v;
      y[rr] = (rr < N) ? c[v] : __builtin_inff();
    }
  }
}

// ---------------------------------------------------------------------------
// Kernel 2a: bitonic sort, global pass for stride j >= 512 (L2-resident: the
// whole 1 MB array sits in the 192 MB L2)
// ---------------------------------------------------------------------------
__global__ __launch_bounds__(256) void bitonic_global(float* __restrict__ d, int j, int k)
{
  const int i = blockIdx.x * blockDim.x + threadIdx.x;
  const int ixj = i ^ j;
  if (ixj > i) {
    const float a = d[i];
    const float b = d[ixj];
    const bool asc = ((i & k) == 0);
    if ((a > b) == asc) { d[i] = b; d[ixj] = a; }
  }
}

// ---------------------------------------------------------------------------
// Kernel 2b: bitonic sort, LDS-resident sweep of all strides jstart..1 (<=256)
// for the current stage k; each block owns 512 contiguous elements.
// ---------------------------------------------------------------------------
__global__ __launch_bounds__(256) void bitonic_local(float* __restrict__ d, int k, int jstart)
{
  __shared__ float s[512];
  const int base = blockIdx.x * 512;
  s[threadIdx.x]       = d[base + threadIdx.x];
  s[threadIdx.x + 256] = d[base + threadIdx.x + 256];
  __syncthreads();
  for (int j = jstart; j >= 1; j >>= 1) {
    const int tix = threadIdx.x;
    // map thread -> smaller index of its compare pair (insert 0 at bit log2(j))
    const int t   = ((tix & ~(j - 1)) << 1) | (tix & (j - 1));
    const int txj = t | j;
    const bool asc = (((base + t) & k) == 0);
    const float a = s[t];
    const float b = s[txj];
    if ((a > b) == asc) { s[t] = b; s[txj] = a; }
    __syncthreads();
  }
  d[base + threadIdx.x]       = s[threadIdx.x];
  d[base + threadIdx.x + 256] = s[threadIdx.x + 256];
}

// ---------------------------------------------------------------------------
// Kernel 3: Hartigan dip statistic on the sorted array (single wave32).
// Faithful port of the diptest C code (min_is_0=False). The GCM/LCM pointer
// builds are inherently sequential (stack algorithm) -> lane 0; the inner
// segment arg-max scans are wave-parallel with shfl_xor fp64 max reductions.
// ---------------------------------------------------------------------------
__global__ __launch_bounds__(32) void dip_kernel(
    const float* __restrict__ xsrt, int N,
    int* __restrict__ mn, int* __restrict__ mjp,
    int* __restrict__ gcm, int* __restrict__ lcm,
    float* __restrict__ out)
{
  const int lane = threadIdx.x;
  auto X = [&](int i) -> double { return (double)xsrt[i - 1]; }; // 1-based

  const double x1 = (double)xsrt[0];
  const double xN = (double)xsrt[N - 1];
  if (N < 4 || x1 == xN) {
    if (lane == 0) out[0] = (float)(1.0 / (2.0 * (double)(N > 1 ? N : 1)));
    return;
  }

  if (lane == 0) {
    // greatest convex minorant pointers mn[1..N]
    mn[1] = 1;
    for (int j = 2; j <= N; ++j) {
      mn[j] = j - 1;
      for (;;) {
        const int mnj = mn[j];
        const int mnmnj = mn[mnj];
        if (mnj == 1 ||
            (X(j) - X(mnj)) * (double)(mnj - mnmnj) <
                (X(mnj) - X(mnmnj)) * (double)(j - mnj))
          break;
        mn[j] = mnmnj;
      }
    }
    // least concave majorant pointers mj[1..N]
    mjp[N] = N;
    for (int k = N - 1; k >= 1; --k) {
      mjp[k] = k + 1;
      for (;;) {
        const int mjk = mjp[k];
        const int mjmjk = mjp[mjk];
        if (mjk == N ||
            (X(k) - X(mjk)) * (double)(mjk - mjmjk) <
                (X(mjk) - X(mjmjk)) * (double)(k - mjk))
          break;
        mjp[k] = mjmjk;
      }
    }
  }
  __syncthreads();

  int low = 1, high = N;
  double dip = 1.0; // in count units; final result dip/(2N)

  for (;;) {
    // -- build gcm (all lanes track 'cur' in registers; lane 0 stores)
    int i = 1, cur = high;
    if (lane == 0) gcm[1] = cur;
    while (cur > low) { cur = mn[cur]; ++i; if (lane == 0) gcm[i] = cur; }
    const int l_gcm = i;
    int ig = i;
    int ix = ig - 1;
    // -- build lcm
    i = 1; cur = low;
    if (lane == 0) lcm[1] = cur;
    while (cur < high) { cur = mjp[cur]; ++i; if (lane == 0) lcm[i] = cur; }
    const int l_lcm = i;
    int ih = i;
    int iv = 2;
    __syncthreads();

    double d = 0.0;
    if (l_gcm != 2 || l_lcm != 2) {
      for (;;) {
        const int gcmix = gcm[ix];
        const int lcmiv = lcm[iv];
        if (gcmix > lcmiv) {
          const int gcmi1 = gcm[ix + 1];
          const double t = ((double)(lcmiv - gcmi1) + 1.0) -
                           (X(lcmiv) - X(gcmi1)) * (double)(gcmix - gcmi1) /
                               (X(gcmix) - X(gcmi1));
          iv += 1;
          if (t >= d) { d = t; ig = ix + 1; ih = iv - 1; }
        } else {
          const int lcmiv1 = lcm[iv - 1];
          const double t = (X(gcmix) - X(lcmiv1)) * (double)(lcmiv - lcmiv1) /
                               (X(lcmiv) - X(lcmiv1)) -
                           ((double)(gcmix - lcmiv1) - 1.0);
          ix -= 1;
          if (t > d) { d = t; ig = ix + 1; ih = iv; }
        }
        if (ix < 1) ix = 1;
        if (iv > l_lcm) iv = l_lcm;
        if (gcm[ix] == lcm[iv]) break;
      }
    } else {
      d = 1.0;
    }
    if (d < dip) break;

    // -- dip of the convex minorant (wave-parallel segment scans)
    double dip_l = 0.0;
    for (int j = ig; j < l_gcm; ++j) {
      const int jb = gcm[j + 1];
      const int je = gcm[j];
      double max_t = 1.0;
      if (je - jb > 1 && X(je) != X(jb)) {
        const double C = (double)(je - jb) / (X(je) - X(jb));
        const double xjb = X(jb);
        double lmax = -1e308;
        for (int t = lane; t <= je - jb; t += 32) {
          const double v = (double)(t + 1) - (X(jb + t) - xjb) * C;
          if (v > lmax) lmax = v;
        }
        for (int off = 16; off >= 1; off >>= 1) {
          const double o = __shfl_xor(lmax, off, 32);
          if (o > lmax) lmax = o;
        }
        if (lmax > max_t) max_t = lmax;
      }
      if (dip_l < max_t) dip_l = max_t;
    }
    // -- dip of the concave majorant
    double dip_u = 0.0;
    for (int j = ih; j < l_lcm; ++j) {
      const int jb = lcm[j];
      const int je = lcm[j + 1];
      double max_t = 1.0;
      if (je - jb > 1 && X(je) != X(jb)) {
        const double C = (double)(je - jb) / (X(je) - X(jb));
        const double xjb = X(jb);
        double lmax = -1e308;
        for (int t = lane; t <= je - jb; t += 32) {
          const double v = (X(jb + t) - xjb) * C - ((double)t - 1.0);
          if (v > lmax) lmax = v;
        }
        for (int off = 16; off >= 1; off >>= 1) {
          const double o = __shfl_xor(lmax, off, 32);
          if (o > lmax) lmax = o;
        }
        if (lmax > max_t) max_t = lmax;
      }
      if (dip_u < max_t) dip_u = max_t;
    }

    const double dip_new = (dip_u > dip_l) ? dip_u : dip_l;
    if (dip < dip_new) dip = dip_new;

    if (low == gcm[ig] && high == lcm[ih]) break;
    low = gcm[ig];
    high = lcm[ih];
    __syncthreads(); // all lanes done reading gcm/lcm before next-round rewrite
  }

  if (lane == 0) out[0] = (float)(dip / (2.0 * (double)N));
}

// ---------------------------------------------------------------------------
// Host-side orchestration (graph-capture safe: only kernel launches on stream)
// ---------------------------------------------------------------------------
extern "C" void kernel_launch(void* const* d_in, const int* in_sizes, int n_in,
                              void* d_out, int out_size, void* d_ws, size_t ws_size,
                              hipStream_t stream) {
  (void)n_in; (void)out_size; (void)ws_size;
  const float* X = (const float*)d_in[0];
  const float* P = (const float*)d_in[1];
  const int* pidx = (const int*)d_in[2];
  float* out = (float*)d_out;

  const int N = in_sizes[0] / DCONST;   // 200000
  int Npad = 512;                       // >=512 so the LDS bitonic kernel applies
  while (Npad < N) Npad <<= 1;          // 262144 for N=200000

  // workspace carving (256B-aligned regions)
  char* ws = (char*)d_ws;
  size_t off = 0;
  auto take = [&](size_t bytes) -> char* {
    char* ptr = ws + off;
    off = (off + bytes + 255) & ~(size_t)255;
    return ptr;
  };
  float* xproj = (float*)take((size_t)Npad * sizeof(float));
  int* mn  = (int*)take((size_t)(N + 2) * sizeof(int));
  int* mj  = (int*)take((size_t)(N + 2) * sizeof(int));
  int* gcm = (int*)take((size_t)(N + 2) * sizeof(int));
  int* lcm = (int*)take((size_t)(N + 2) * sizeof(int));

  // 1) projection matvec (WMMA fp32), +INF padding to Npad
  matvec_wmma<<<Npad / 64, 128, 0, stream>>>(X, P, pidx, xproj, N, Npad);

  // 2) bitonic sort ascending over Npad elements (all traffic L2-resident)
  for (int k = 2; k <= Npad; k <<= 1) {
    int j = k >> 1;
    for (; j >= 512; j >>= 1)
      bitonic_global<<<Npad / 256, 256, 0, stream>>>(xproj, j, k);
    bitonic_local<<<Npad / 512, 256, 0, stream>>>(xproj, k, j);
  }

  // 3) dip statistic on sorted prefix [0, N)
  dip_kernel<<<1, 32, 0, stream>>>(xproj, N, mn, mj, gcm, lcm, out);
}